// GraphAttentionConv_54700703481982
// MI455X (gfx1250) — compile-verified
//
#include <hip/hip_runtime.h>
#include <hip/hip_bf16.h>

// Problem constants (fixed by the reference).
#define Bb   8
#define Nn   2048
#define CIN  128
#define COUT 64
#define Mrows (Bb * Nn)   // 16384

typedef __attribute__((ext_vector_type(2))) float v2f;
typedef __attribute__((ext_vector_type(8))) float v8f;

// ---------------------------------------------------------------------------
// Pass 1: h = x @ W^T using V_WMMA_F32_16X16X4_F32 (f32 in / f32 acc, exact).
// One wave (32 threads) computes one 16x16 tile of h; 32 chained WMMAs over
// K=128. Grid = (M/16, COUT/16) = (1024, 4).
//
// A-frag (16x4 f32): lanes 0-15 -> K = k+0,k+1 ; lanes 16-31 -> K = k+2,k+3.
// B-frag (4x16 f32): same K split; N = lane & 15. B[k][n] = W[n*CIN + k],
// so two consecutive K are contiguous in a W row -> v2f loads.
// C/D (16x16 f32): VGPR r -> row r (lanes 0-15) / row r+8 (lanes 16-31).
// ---------------------------------------------------------------------------
__global__ __launch_bounds__(32) void gat_h_wmma(const float* __restrict__ x,
                                                 const float* __restrict__ W,
                                                 float* __restrict__ h) {
  const int lane  = threadIdx.x;        // wave32: block == one wave
  const int lmod  = lane & 15;
  const int khalf = (lane >> 4) << 1;   // 0 or 2
  const int row0  = blockIdx.x << 4;
  const int col0  = blockIdx.y << 4;

  const float* xa = x + (size_t)(row0 + lmod) * CIN + khalf; // A: row = M
  const float* wb = W + (size_t)(col0 + lmod) * CIN + khalf; // B: row = N (W^T)

  v8f acc = {};
#pragma unroll
  for (int k = 0; k < CIN; k += 4) {
    v2f a = *(const v2f*)(xa + k);
    v2f b = *(const v2f*)(wb + k);
    // 8 args: (neg_a, A, neg_b, B, c_mod, C, reuse_a, reuse_b)
    acc = __builtin_amdgcn_wmma_f32_16x16x4_f32(false, a, false, b,
                                                (short)0, acc, false, false);
  }

  const int mbase = row0 + ((lane >> 4) << 3); // rows 0-7 or 8-15 of the tile
  float* hp = h + (size_t)mbase * COUT + col0 + lmod;
#pragma unroll
  for (int r = 0; r < 8; ++r) hp[(size_t)r * COUT] = acc[r];
}

// ---------------------------------------------------------------------------
// Pass 2: e[i] = exp(h[i,:] . a_dst).  (a_src cancels in the row softmax.)
// One thread per node; 4 MB of h read -> negligible.
// ---------------------------------------------------------------------------
__global__ __launch_bounds__(256) void gat_e(const float* __restrict__ h,
                                             const float* __restrict__ a_w,
                                             float* __restrict__ e) {
  const int i = blockIdx.x * blockDim.x + threadIdx.x;
  if (i >= Mrows) return;
  const float*  hr   = h + (size_t)i * COUT;
  const float*  adst = a_w + COUT;     // a_w is (1, 2*COUT): [a_src | a_dst]
  float s = 0.f;
#pragma unroll
  for (int c = 0; c < COUT; c += 4) {
    float4 hv = *(const float4*)(hr + c);
    float4 av = *(const float4*)(adst + c);
    s += hv.x * av.x + hv.y * av.y + hv.z * av.z + hv.w * av.w;
  }
  s = fminf(fmaxf(s, -60.f), 60.f);    // constant shifts cancel in the ratio
  e[i] = __expf(s);
}

// ---------------------------------------------------------------------------
// Pass 3 (dominant, streams 134 MB of adj):
//   denom[b,i] = sum_j adj[b,i,j] * e[b,j]
//   out[b,i,c] = relu( (e[b,i]/denom) * h[b,i,c] + bias[c] )
// e[b,:] (8 KB) is staged in LDS; each wave owns 2 rows, reads adj with
// float4 (b128) loads, wave-reduces with shfl_xor (wave32).
// Grid = (Nn/16, Bb), 256 threads (8 waves).
// ---------------------------------------------------------------------------
__global__ __launch_bounds__(256) void gat_out(const float* __restrict__ adj,
                                               const float* __restrict__ h,
                                               const float* __restrict__ e,
                                               const float* __restrict__ bias,
                                               float* __restrict__ out) {
  __shared__ __align__(16) float se[Nn];
  const int b   = blockIdx.y;
  const int tid = threadIdx.x;

  const float* eb = e + (size_t)b * Nn;
  for (int j = tid; j < Nn; j += 256) se[j] = eb[j];
  __syncthreads();

  const float4* se4 = (const float4*)se;
  const int wave = tid >> 5;
  const int lane = tid & 31;
  const int row_base = blockIdx.x * 16;   // 16 rows per block, 2 per wave

#pragma unroll
  for (int rr = 0; rr < 2; ++rr) {
    const int i = row_base + wave * 2 + rr;
    const float4* arow = (const float4*)(adj + ((size_t)b * Nn + i) * Nn);

    float acc = 0.f;
#pragma unroll
    for (int it = 0; it < Nn / 128; ++it) {      // 16 iterations
      const int j4 = it * 32 + lane;
      float4 a  = arow[j4];
      float4 ev = se4[j4];
      acc += a.x * ev.x + a.y * ev.y + a.z * ev.z + a.w * ev.w;
    }
    // wave32 all-reduce
#pragma unroll
    for (int off = 16; off > 0; off >>= 1)
      acc += __shfl_xor(acc, off, 32);

    const float w = se[i] / acc;                 // attn[i,i]

    const int c = lane * 2;                      // 64 outputs, 2 per lane
    const float* hrow = h   + ((size_t)b * Nn + i) * COUT;
    float*       orow = out + ((size_t)b * Nn + i) * COUT;
    float2 hv = *(const float2*)(hrow + c);
    float2 ov;
    ov.x = fmaxf(w * hv.x + bias[c],     0.f);
    ov.y = fmaxf(w * hv.y + bias[c + 1], 0.f);
    *(float2*)(orow + c) = ov;
  }
}

// ---------------------------------------------------------------------------
extern "C" void kernel_launch(void* const* d_in, const int* in_sizes, int n_in,
                              void* d_out, int out_size, void* d_ws, size_t ws_size,
                              hipStream_t stream) {
  const float* x    = (const float*)d_in[0];  // (8, 2048, 128)
  const float* adj  = (const float*)d_in[1];  // (8, 2048, 2048)
  const float* W_w  = (const float*)d_in[2];  // (64, 128)
  const float* a_w  = (const float*)d_in[3];  // (1, 128)
  const float* bias = (const float*)d_in[4];  // (64,)
  float* out = (float*)d_out;                 // (8, 2048, 64)

  float* h = (float*)d_ws;                    // Mrows * COUT floats (4 MB)
  float* e = h + (size_t)Mrows * COUT;        // Mrows floats (64 KB)

  gat_h_wmma<<<dim3(Mrows / 16, COUT / 16), 32, 0, stream>>>(x, W_w, h);
  gat_e<<<Mrows / 256, 256, 0, stream>>>(h, a_w, e);
  gat_out<<<dim3(Nn / 16, Bb), 256, 0, stream>>>(adj, h, e, bias, out);
}